// GCN_2276332667312
// MI455X (gfx1250) — compile-verified
//
#include <hip/hip_runtime.h>
#include <hip/hip_bf16.h>

typedef __attribute__((ext_vector_type(2))) float v2f;
typedef __attribute__((ext_vector_type(8))) float v8f;

#define IN_FEATS 128
#define H_FEATS  64
#define N_GRAPHS 256

// ---------------- degree / norm kernels ----------------

__global__ void init_deg_kernel(float* __restrict__ odeg, float* __restrict__ ideg, int n) {
    int i = blockIdx.x * blockDim.x + threadIdx.x;
    if (i < n) { odeg[i] = 1.0f; ideg[i] = 1.0f; }   // self-loop contributes 1 to both
}

__global__ void deg_accum_kernel(const int* __restrict__ src, const int* __restrict__ dst,
                                 float* __restrict__ odeg, float* __restrict__ ideg, int n_edges) {
    int e = blockIdx.x * blockDim.x + threadIdx.x;
    if (e < n_edges) {
        atomicAdd(&odeg[src[e]], 1.0f);
        atomicAdd(&ideg[dst[e]], 1.0f);
    }
}

__global__ void rsqrt_kernel(float* __restrict__ odeg, float* __restrict__ ideg, int n) {
    int i = blockIdx.x * blockDim.x + threadIdx.x;
    if (i < n) { odeg[i] = rsqrtf(odeg[i]); ideg[i] = rsqrtf(ideg[i]); }
}

// ---------------- GEMM1: h1pre = (x * norm_src) @ W1  (M=100000,K=128,N=64) ------------
// One wave computes a 16x64 row block using V_WMMA_F32_16X16X4_F32.
// A 16x4 f32 fragment: lanes 0-15 -> K=k0,k0+1 ; lanes 16-31 -> K=k0+2,k0+3 (2 VGPRs).
// C 16x16 f32: VGPR r -> M=r (lanes 0-15) / M=8+r (lanes 16-31), N = lane%16.

__global__ __launch_bounds__(128) void gemm1_wmma_kernel(
    const float* __restrict__ x, const float* __restrict__ W1,
    const float* __restrict__ norm_src, float* __restrict__ h1pre, int n_nodes) {

    const int wave  = blockIdx.x * (blockDim.x >> 5) + (threadIdx.x >> 5);
    const int lane  = threadIdx.x & 31;
    const int mbase = wave * 16;
    if (mbase >= n_nodes) return;

    const int hl   = lane & 15;          // lane within half-wave
    const int koff = (lane >> 4) << 1;   // 0 for lanes 0-15, 2 for lanes 16-31

    const int   row  = mbase + hl;       // A-fragment source row for this lane
    const float nrm  = norm_src[row];
    const float* xrow = x + (long)row * IN_FEATS;

    v8f acc0 = {}, acc1 = {}, acc2 = {}, acc3 = {};

    #pragma unroll 4
    for (int k0 = 0; k0 < IN_FEATS; k0 += 4) {
        v2f a;
        a.x = xrow[k0 + koff]     * nrm;
        a.y = xrow[k0 + koff + 1] * nrm;

        const float* wr0 = W1 + (long)(k0 + koff) * H_FEATS;  // row k0+koff of W1
        const float* wr1 = wr0 + H_FEATS;                     // row k0+koff+1
        v2f b0, b1v, b2v, b3v;
        b0.x  = wr0[hl];      b0.y  = wr1[hl];
        b1v.x = wr0[16 + hl]; b1v.y = wr1[16 + hl];
        b2v.x = wr0[32 + hl]; b2v.y = wr1[32 + hl];
        b3v.x = wr0[48 + hl]; b3v.y = wr1[48 + hl];

        acc0 = __builtin_amdgcn_wmma_f32_16x16x4_f32(false, a, false, b0,  (short)0, acc0, false, false);
        acc1 = __builtin_amdgcn_wmma_f32_16x16x4_f32(false, a, false, b1v, (short)0, acc1, false, false);
        acc2 = __builtin_amdgcn_wmma_f32_16x16x4_f32(false, a, false, b2v, (short)0, acc2, false, false);
        acc3 = __builtin_amdgcn_wmma_f32_16x16x4_f32(false, a, false, b3v, (short)0, acc3, false, false);
    }

    // store D: row = mbase + r + (lane>=16 ? 8 : 0), col = ncol + hl
    const int srow = mbase + ((lane >> 4) << 3);
    float* out = h1pre + (long)srow * H_FEATS + hl;
    #pragma unroll
    for (int r = 0; r < 8; ++r) {
        out[(long)r * H_FEATS + 0]  = acc0[r];
        out[(long)r * H_FEATS + 16] = acc1[r];
        out[(long)r * H_FEATS + 32] = acc2[r];
        out[(long)r * H_FEATS + 48] = acc3[r];
    }
}

// ---------------- edge scatter (layer 1): agg1[dst] += h1pre[src], 64 feats/edge ------

__global__ void scatter1_kernel(const int* __restrict__ src, const int* __restrict__ dst,
                                const float* __restrict__ h1pre, float* __restrict__ agg1,
                                int n_edges) {
    long idx = (long)blockIdx.x * blockDim.x + threadIdx.x;
    long e = idx >> 6;
    int  f = (int)(idx & 63);
    if (e >= n_edges) return;
    int s = src[e], d = dst[e];
    atomicAdd(&agg1[(long)d * H_FEATS + f], h1pre[(long)s * H_FEATS + f]);
}

// combine1: h1s = relu((agg1 + h1pre_selfloop) * norm_dst + b1) * norm_src   (in place in agg1)

__global__ void combine1_kernel(float* __restrict__ agg1, const float* __restrict__ h1pre,
                                const float* __restrict__ norm_dst, const float* __restrict__ norm_src,
                                const float* __restrict__ b1, int n_nodes) {
    long idx = (long)blockIdx.x * blockDim.x + threadIdx.x;
    if (idx >= (long)n_nodes * H_FEATS) return;
    long i = idx >> 6;
    int  f = (int)(idx & 63);
    float v = (agg1[idx] + h1pre[idx]) * norm_dst[i] + b1[f];
    v = fmaxf(v, 0.0f);
    agg1[idx] = v * norm_src[i];
}

// ---------------- GEMM2: h2pre = h1s @ W2  (64 -> 2) ----------------

__global__ __launch_bounds__(256) void gemm2_kernel(const float* __restrict__ h1s,
                                                    const float* __restrict__ W2,
                                                    float* __restrict__ h2pre, int n_nodes) {
    __shared__ float w[H_FEATS * 2];
    if (threadIdx.x < H_FEATS * 2) w[threadIdx.x] = W2[threadIdx.x];
    __syncthreads();
    int i = blockIdx.x * blockDim.x + threadIdx.x;
    if (i >= n_nodes) return;
    const float* row = h1s + (long)i * H_FEATS;
    float c0 = 0.0f, c1 = 0.0f;
    #pragma unroll
    for (int k = 0; k < H_FEATS; ++k) {
        float h = row[k];
        c0 += h * w[2 * k];
        c1 += h * w[2 * k + 1];
    }
    h2pre[2L * i]     = c0;
    h2pre[2L * i + 1] = c1;
}

// ---------------- edge scatter (layer 2): agg2[dst] += h2pre[src], 2 feats/edge -------

__global__ void scatter2_kernel(const int* __restrict__ src, const int* __restrict__ dst,
                                const float* __restrict__ h2pre, float* __restrict__ agg2,
                                int n_edges) {
    long idx = (long)blockIdx.x * blockDim.x + threadIdx.x;
    long e = idx >> 1;
    int  c = (int)(idx & 1);
    if (e >= n_edges) return;
    int s = src[e], d = dst[e];
    atomicAdd(&agg2[2L * d + c], h2pre[2L * s + c]);
}

// combine2 + graph mean-pool accumulation

__global__ void combine2_pool_kernel(const float* __restrict__ agg2, const float* __restrict__ h2pre,
                                     const float* __restrict__ norm_dst, const float* __restrict__ b2,
                                     const int* __restrict__ graph_id,
                                     float* __restrict__ sums, float* __restrict__ cnts, int n_nodes) {
    long idx = (long)blockIdx.x * blockDim.x + threadIdx.x;
    if (idx >= 2L * n_nodes) return;
    long i = idx >> 1;
    int  c = (int)(idx & 1);
    float v = (agg2[idx] + h2pre[idx]) * norm_dst[i] + b2[c];
    int g = graph_id[i];
    atomicAdd(&sums[2 * g + c], v);
    if (c == 0) atomicAdd(&cnts[g], 1.0f);
}

__global__ void finalize_kernel(const float* __restrict__ sums, const float* __restrict__ cnts,
                                float* __restrict__ out) {
    int idx = blockIdx.x * blockDim.x + threadIdx.x;
    if (idx < 2 * N_GRAPHS) out[idx] = sums[idx] / fmaxf(cnts[idx >> 1], 1.0f);
}

// ---------------- launch ----------------

extern "C" void kernel_launch(void* const* d_in, const int* in_sizes, int n_in,
                              void* d_out, int out_size, void* d_ws, size_t ws_size,
                              hipStream_t stream) {
    const float* x   = (const float*)d_in[0];
    const float* W1  = (const float*)d_in[1];
    const float* b1  = (const float*)d_in[2];
    const float* W2  = (const float*)d_in[3];
    const float* b2  = (const float*)d_in[4];
    const int*   src = (const int*)d_in[5];
    const int*   dst = (const int*)d_in[6];
    const int*   gid = (const int*)d_in[7];

    const int n_nodes = in_sizes[0] / IN_FEATS;   // 100000
    const int n_edges = in_sizes[5];              // 1600000

    float* ws    = (float*)d_ws;
    float* h1pre = ws;                                  // n*64
    float* agg1  = h1pre + (size_t)n_nodes * H_FEATS;   // n*64
    float* h2pre = agg1  + (size_t)n_nodes * H_FEATS;   // n*2
    float* agg2  = h2pre + (size_t)n_nodes * 2;         // n*2
    float* odeg  = agg2  + (size_t)n_nodes * 2;         // n
    float* ideg  = odeg  + (size_t)n_nodes;             // n
    float* sums  = ideg  + (size_t)n_nodes;             // 512
    float* cnts  = sums  + 2 * N_GRAPHS;                // 256

    // zero accumulators (ws is poisoned once and never re-poisoned; do this every call)
    hipMemsetAsync(agg1, 0, (size_t)n_nodes * H_FEATS * sizeof(float), stream);
    hipMemsetAsync(agg2, 0, (size_t)n_nodes * 2 * sizeof(float), stream);
    hipMemsetAsync(sums, 0, (2 * N_GRAPHS + N_GRAPHS) * sizeof(float), stream);

    // degrees -> norms
    init_deg_kernel<<<(n_nodes + 255) / 256, 256, 0, stream>>>(odeg, ideg, n_nodes);
    deg_accum_kernel<<<(n_edges + 255) / 256, 256, 0, stream>>>(src, dst, odeg, ideg, n_edges);
    rsqrt_kernel<<<(n_nodes + 255) / 256, 256, 0, stream>>>(odeg, ideg, n_nodes);

    // layer 1
    const int mtiles = (n_nodes + 15) / 16;           // 6250 waves
    gemm1_wmma_kernel<<<(mtiles + 3) / 4, 128, 0, stream>>>(x, W1, odeg, h1pre, n_nodes);
    {
        long total = (long)n_edges * H_FEATS;
        scatter1_kernel<<<(unsigned)((total + 255) / 256), 256, 0, stream>>>(src, dst, h1pre, agg1, n_edges);
    }
    {
        long total = (long)n_nodes * H_FEATS;
        combine1_kernel<<<(unsigned)((total + 255) / 256), 256, 0, stream>>>(agg1, h1pre, ideg, odeg, b1, n_nodes);
    }

    // layer 2 (agg1 now holds h1s = relu(...)*norm_src)
    gemm2_kernel<<<(n_nodes + 255) / 256, 256, 0, stream>>>(agg1, W2, h2pre, n_nodes);
    {
        long total = 2L * n_edges;
        scatter2_kernel<<<(unsigned)((total + 255) / 256), 256, 0, stream>>>(src, dst, h2pre, agg2, n_edges);
    }
    {
        long total = 2L * n_nodes;
        combine2_pool_kernel<<<(unsigned)((total + 255) / 256), 256, 0, stream>>>(
            agg2, h2pre, ideg, b2, gid, sums, cnts, n_nodes);
    }

    finalize_kernel<<<2, 256, 0, stream>>>(sums, cnts, (float*)d_out);
}